// CosinePrediction_55035710931253
// MI455X (gfx1250) — compile-verified
//
#include <hip/hip_runtime.h>
#include <hip/hip_bf16.h>

typedef __attribute__((ext_vector_type(2))) float v2f;
typedef __attribute__((ext_vector_type(8))) float v8f;

#define L2EPS 1e-12f
#define DIM 64

// --- Pass 1: per-row reciprocal L2 norm: rnorm[row] = 1 / max(||x||, eps) ---
// One wave32 per 64-float row; each lane loads a float2, shfl-reduce.
__global__ __launch_bounds__(256) void CosPred_norm_kernel(
    const float* __restrict__ h_user, const float* __restrict__ h_item,
    int nU, int nI, float* __restrict__ rnorm) {
  int wave = (int)((blockIdx.x * blockDim.x + threadIdx.x) >> 5);
  int lane = (int)(threadIdx.x & 31);
  int total = nU + nI;
  if (wave >= total) return;
  const float* row = (wave < nU) ? (h_user + (size_t)wave * DIM)
                                 : (h_item + (size_t)(wave - nU) * DIM);
  v2f v = *(const v2f*)(row + lane * 2);
  float s = v.x * v.x + v.y * v.y;
#pragma unroll
  for (int off = 16; off >= 1; off >>= 1)
    s += __shfl_xor(s, off, 32);
  if (lane == 0)
    rnorm[wave] = 1.0f / fmaxf(sqrtf(s), L2EPS);
}

// --- Pass 2: 16 edges per wave via V_WMMA_F32_16X16X4_F32 diagonal ---
// A = h_user rows of the 16 src indices (16xK), B = h_item rows of the 16 dst
// indices as KxN columns. Accumulate over 16 K-tiles (K=64), take diag,
// scale by rnorm_u[src]*rnorm_i[dst].
__global__ __launch_bounds__(256) void CosPred_edge_wmma_kernel(
    const float* __restrict__ h_user, const float* __restrict__ h_item,
    const int* __restrict__ src_idx, const int* __restrict__ dst_idx,
    const float* __restrict__ rnU, const float* __restrict__ rnI,
    float* __restrict__ out, int E) {
  int gwave = (int)((blockIdx.x * blockDim.x + threadIdx.x) >> 5);  // wave-uniform
  int lane  = (int)(threadIdx.x & 31);
  int eBase = gwave * 16;
  if (eBase >= E) return;  // uniform across the wave -> EXEC all-1s below

  // Lane L handles edge eBase + (L & 15); upper half-wave covers K offsets +2,+3.
  int myE = eBase + (lane & 15);
  int eClamped = min(myE, E - 1);           // clamp loads; stores predicated later
  int s = src_idx[eClamped];
  int d = dst_idx[eClamped];
  int kOff = (lane >> 4) << 1;              // 0 for lanes 0-15, 2 for lanes 16-31

  const float* aRow = h_user + (size_t)s * DIM + kOff;
  const float* bRow = h_item + (size_t)d * DIM + kOff;

  v8f c = {};
#pragma unroll
  for (int kt = 0; kt < DIM / 4; ++kt) {
    v2f a = *(const v2f*)(aRow + kt * 4);   // A[M=lane&15][K=kt*4+kOff .. +1]
    v2f b = *(const v2f*)(bRow + kt * 4);   // B[K=kt*4+kOff .. +1][N=lane&15]
    c = __builtin_amdgcn_wmma_f32_16x16x4_f32(
        /*neg_a=*/false, a, /*neg_b=*/false, b,
        /*c_mod=*/(short)0, c, /*reuse_a=*/false, /*reuse_b=*/false);
  }

  // cos[e] = (u . v) * (1/max(|u|,eps)) * (1/max(|v|,eps))
  float scale = rnU[s] * rnI[d];

  // Diagonal extraction: D[r][r] -> VGPR r lane r ; D[8+r][8+r] -> VGPR r lane 24+r.
#pragma unroll
  for (int r = 0; r < 8; ++r) {
    if ((lane == r) || (lane == 24 + r)) {
      if (myE < E) out[myE] = c[r] * scale;
    }
  }
}

extern "C" void kernel_launch(void* const* d_in, const int* in_sizes, int n_in,
                              void* d_out, int out_size, void* d_ws, size_t ws_size,
                              hipStream_t stream) {
  const float* h_user = (const float*)d_in[0];
  const float* h_item = (const float*)d_in[1];
  const int*   src    = (const int*)d_in[2];
  const int*   dst    = (const int*)d_in[3];
  float*       out    = (float*)d_out;

  int nU = in_sizes[0] / DIM;
  int nI = in_sizes[1] / DIM;
  int E  = in_sizes[2];

  float* rnorm = (float*)d_ws;  // nU + nI floats (~800 KB)

  // Pass 1: norms. One wave per row, 8 rows per 256-thread block.
  {
    long long totalRows = (long long)nU + nI;
    long long totalThreads = totalRows * 32;
    int blocks = (int)((totalThreads + 255) / 256);
    CosPred_norm_kernel<<<blocks, 256, 0, stream>>>(h_user, h_item, nU, nI, rnorm);
  }

  // Pass 2: edge dots via WMMA. 16 edges per wave, 8 waves per block.
  {
    int numWaves = (E + 15) / 16;
    int blocks = (numWaves + 7) / 8;
    CosPred_edge_wmma_kernel<<<blocks, 256, 0, stream>>>(
        h_user, h_item, src, dst, rnorm, rnorm + nU, out, E);
  }
}